// _GS_model_67843303408132
// MI455X (gfx1250) — compile-verified
//
#include <hip/hip_runtime.h>
#include <hip/hip_bf16.h>

#define EPSF 1e-8f
#define SORT_TB 256
#define SORT_IPT 4
#define SORT_TILE (SORT_TB * SORT_IPT)

typedef float v2f __attribute__((ext_vector_type(2)));
typedef float v8f __attribute__((ext_vector_type(8)));

// ---------------------------------------------------------------------------
// Fused per-splat compute. One wave = one tile of 16 gaussians.
// WMMA f32 16x16x4: A1 = stacked P (4 cams x 3 rows), A2 = stacked K*P.
// D1 = camera-space xyz, D2 = pixel-homogeneous xyz for 16 gaussians x 4 cams.
// ---------------------------------------------------------------------------
__global__ __launch_bounds__(256)
void gs_compute_kernel(const float* __restrict__ mean,
                       const float* __restrict__ vq,
                       const float* __restrict__ vs,
                       const float* __restrict__ opa,
                       const float* __restrict__ color,
                       const float* __restrict__ P,
                       const float* __restrict__ K,
                       float* __restrict__ unsorted,
                       unsigned* __restrict__ keys,
                       unsigned* __restrict__ idx,
                       int n, int numTiles16)
{
    __shared__ float stage[8][2][16][17];   // [wave][mat][row M][col j], pad 17
    const int tid  = threadIdx.x;
    const int lane = tid & 31;
    const int wave = tid >> 5;
    const int tile = blockIdx.x * 8 + wave;
    if (tile >= numTiles16) return;         // wave-uniform guard: EXEC all-1s below
    const int colId = lane & 15;            // gaussian within tile / A row
    const int kk    = (lane >> 4) * 2;      // K columns held by this lane

    // ---- A fragments (wave-uniform matrices, per-lane fragment) ----
    v2f a1 = {0.f, 0.f}, a2 = {0.f, 0.f};
    if (colId < 12) {
        int cb = colId / 3, ci = colId % 3;
        const float* Pb = P + cb * 12;
        a1.x = Pb[ci*4 + kk];
        a1.y = Pb[ci*4 + kk + 1];
        const float* Kb = K + cb * 9 + ci * 3;
        a2.x = Kb[0]*Pb[0*4+kk]   + Kb[1]*Pb[1*4+kk]   + Kb[2]*Pb[2*4+kk];
        a2.y = Kb[0]*Pb[0*4+kk+1] + Kb[1]*Pb[1*4+kk+1] + Kb[2]*Pb[2*4+kk+1];
    }

    // ---- B fragment: homogeneous positions of 16 gaussians ----
    const int g  = tile * 16 + colId;
    const int gc = (g < n) ? g : (n - 1);
    v2f bm;
    if (lane < 16) { bm.x = mean[gc*3+0]; bm.y = mean[gc*3+1]; }   // K rows 0,1
    else           { bm.x = mean[gc*3+2]; bm.y = 1.0f;        }   // K rows 2,3

    v8f cz = {0.f,0.f,0.f,0.f,0.f,0.f,0.f,0.f};
    v8f d1 = __builtin_amdgcn_wmma_f32_16x16x4_f32(false, a1, false, bm, (short)0, cz, false, false);
    v8f d2 = __builtin_amdgcn_wmma_f32_16x16x4_f32(false, a2, false, bm, (short)0, cz, false, false);

    const int rowOff = (lane >> 4) * 8;     // D layout: VGPR r -> row r (+8 for hi lanes)
    #pragma unroll
    for (int r = 0; r < 8; r++) {
        stage[wave][0][rowOff + r][colId] = d1[r];
        stage[wave][1][rowOff + r][colId] = d2[r];
    }

    // ---- Per-gaussian (camera independent) work ----
    float qw = vq[gc*4+0], qx = vq[gc*4+1], qy = vq[gc*4+2], qz = vq[gc*4+3];
    float qn = fmaxf(sqrtf(qw*qw + qx*qx + qy*qy + qz*qz), EPSF);
    qw /= qn; qx /= qn; qy /= qn; qz /= qn;
    float r00 = 1.f - 2.f*(qy*qy + qz*qz), r01 = 2.f*(qx*qy - qw*qz), r02 = 2.f*(qx*qz + qw*qy);
    float r10 = 2.f*(qx*qy + qw*qz), r11 = 1.f - 2.f*(qx*qx + qz*qz), r12 = 2.f*(qy*qz - qw*qx);
    float r20 = 2.f*(qx*qz - qw*qy), r21 = 2.f*(qy*qz + qw*qx), r22 = 1.f - 2.f*(qx*qx + qy*qy);
    float e0 = expf(vs[gc*3+0]), e1 = expf(vs[gc*3+1]), e2 = expf(vs[gc*3+2]);
    float m00=r00*e0, m01=r01*e1, m02=r02*e2;
    float m10=r10*e0, m11=r11*e1, m12=r12*e2;
    float m20=r20*e0, m21=r21*e1, m22=r22*e2;
    float Sxx = m00*m00 + m01*m01 + m02*m02;
    float Sxy = m00*m10 + m01*m11 + m02*m12;
    float Sxz = m00*m20 + m01*m21 + m02*m22;
    float Syy = m10*m10 + m11*m11 + m12*m12;
    float Syz = m10*m20 + m11*m21 + m12*m22;
    float Szz = m20*m20 + m21*m21 + m22*m22;
    float sig = 1.f / (1.f + expf(-opa[gc]));
    const float* col = color + (size_t)gc * 48;
    const bool gValid = (g < n);

    // ---- Two (gaussian, camera) tasks per lane: covers 16 g x 4 cams per wave ----
    #pragma unroll
    for (int t = 0; t < 2; t++) {
        int bcam = (lane >> 4) + 2*t;
        float mc0 = stage[wave][0][3*bcam+0][colId];
        float mc1 = stage[wave][0][3*bcam+1][colId];
        float mc2 = stage[wave][0][3*bcam+2][colId];
        float ph0 = stage[wave][1][3*bcam+0][colId];
        float ph1 = stage[wave][1][3*bcam+1][colId];
        float ph2 = stage[wave][1][3*bcam+2][colId];

        // depth key (unclipped z) -> order-preserving u32
        unsigned zb  = __float_as_uint(mc2);
        unsigned key = zb ^ ((unsigned)((int)zb >> 31) | 0x80000000u);

        float pz = fmaxf(ph2, EPSF);
        float px = ph0 / pz, py = ph1 / pz;

        float fx = K[bcam*9 + 0], fy = K[bcam*9 + 4];
        const float* Pb = P + bcam * 12;
        float R00=Pb[0], R01=Pb[1], R02=Pb[2];
        float R10=Pb[4], R11=Pb[5], R12=Pb[6];
        float R20=Pb[8], R21=Pb[9], R22=Pb[10];
        // T = R * Sigma ; Vc = T * R^T
        float T00 = R00*Sxx + R01*Sxy + R02*Sxz;
        float T01 = R00*Sxy + R01*Syy + R02*Syz;
        float T02 = R00*Sxz + R01*Syz + R02*Szz;
        float T10 = R10*Sxx + R11*Sxy + R12*Sxz;
        float T11 = R10*Sxy + R11*Syy + R12*Syz;
        float T12 = R10*Sxz + R11*Syz + R12*Szz;
        float T20 = R20*Sxx + R21*Sxy + R22*Sxz;
        float T21 = R20*Sxy + R21*Syy + R22*Syz;
        float T22 = R20*Sxz + R21*Syz + R22*Szz;
        float V00 = T00*R00 + T01*R01 + T02*R02;
        float V01 = T00*R10 + T01*R11 + T02*R12;
        float V02 = T00*R20 + T01*R21 + T02*R22;
        float V10 = T10*R00 + T11*R01 + T12*R02;
        float V11 = T10*R10 + T11*R11 + T12*R12;
        float V12 = T10*R20 + T11*R21 + T12*R22;
        float V20 = T20*R00 + T21*R01 + T22*R02;
        float V21 = T20*R10 + T21*R11 + T22*R12;
        float V22 = T20*R20 + T21*R21 + T22*R22;

        float zc  = fmaxf(mc2, EPSF);
        float izc = 1.f / zc;
        float J00 = fx*izc, J02 = -fx*mc0*izc*izc;
        float J11 = fy*izc, J12 = -fy*mc1*izc*izc;
        float u0 = V00*J00 + V02*J02;
        float u1 = V10*J00 + V12*J02;
        float u2 = V20*J00 + V22*J02;
        float w0 = V01*J11 + V02*J12;
        float w1 = V11*J11 + V12*J12;
        float w2 = V21*J11 + V22*J12;
        float Ap = J00*u0 + J02*u2;   // S[0,0]
        float Bp = J00*w0 + J02*w2;   // S[0,1]
        float Cp = J11*u1 + J12*u2;   // S[1,0]
        float Dp = J11*w1 + J12*w2;   // S[1,1]

        // 2x2 symmetric eigen (lower triangle, eigh-style); order-independent boxsize
        float aE = Ap, bE = Cp, cE = Dp;
        float mE = 0.5f*(aE + cE), hd = 0.5f*(aE - cE);
        float rE = sqrtf(hd*hd + bE*bE);
        float lam1 = mE - rE, lam2 = mE + rE;
        float vx = bE, vy = lam2 - aE;
        float n2 = vx*vx + vy*vy;
        float c2 = (n2 > 1e-30f) ? (vx*vx)/n2 : 1.0f;
        float s2 = 1.0f - c2;
        float al1 = fabsf(lam1), al2 = fabsf(lam2);
        float bx = 3.f*sqrtf(s2*al1 + c2*al2);
        float by = 3.f*sqrtf(c2*al1 + s2*al2);

        float det = Ap*Dp - Bp*Cp;
        det = (fabsf(det) < 1e-12f) ? 1e-12f : det;
        float idet = 1.f/det;
        float i00 =  Dp*idet, i01 = -Bp*idet, i10 = -Cp*idet, i11 = Ap*idet;

        // SH basis on -mean_camera
        float dx = -mc0, dy = -mc1, dz = -mc2;
        float dn = fmaxf(sqrtf(dx*dx + dy*dy + dz*dz), EPSF);
        dx /= dn; dy /= dn; dz /= dn;
        float xx = dx*dx, yy = dy*dy, zz = dz*dz;
        float gam[16];
        gam[0]  =  0.28209479177387814f;
        gam[1]  = -0.4886025119029199f*dy;
        gam[2]  =  0.4886025119029199f*dz;
        gam[3]  = -0.4886025119029199f*dx;
        gam[4]  =  1.0925484305920792f*dx*dy;
        gam[5]  = -1.0925484305920792f*dy*dz;
        gam[6]  =  0.31539156525252005f*(2.f*zz - xx - yy);
        gam[7]  = -1.0925484305920792f*dx*dz;
        gam[8]  =  0.5462742152960396f*(xx - yy);
        gam[9]  = -0.5900435899266435f*dy*(3.f*xx - yy);
        gam[10] =  2.890611442640554f*dx*dy*dz;
        gam[11] = -0.4570457994644658f*dy*(4.f*zz - xx - yy);
        gam[12] =  0.3731763325901154f*dz*(2.f*zz - 3.f*xx - 3.f*yy);
        gam[13] = -0.4570457994644658f*dx*(4.f*zz - xx - yy);
        gam[14] =  1.445305721320277f*dz*(xx - yy);
        gam[15] = -0.5900435899266435f*dx*(xx - 3.f*yy);
        float Ld0 = 0.f, Ld1 = 0.f, Ld2 = 0.f;
        #pragma unroll
        for (int kq = 0; kq < 16; kq++) {
            float gk = gam[kq];
            Ld0 += gk * col[kq*3+0];
            Ld1 += gk * col[kq*3+1];
            Ld2 += gk * col[kq*3+2];
        }

        if (gValid) {
            size_t oo = ((size_t)bcam * n + g) * 12;
            unsorted[oo+0]=px;  unsorted[oo+1]=py;  unsorted[oo+2]=sig;
            unsorted[oo+3]=i00; unsorted[oo+4]=i01; unsorted[oo+5]=i10; unsorted[oo+6]=i11;
            unsorted[oo+7]=Ld0; unsorted[oo+8]=Ld1; unsorted[oo+9]=Ld2;
            unsorted[oo+10]=bx; unsorted[oo+11]=by;
            keys[(size_t)bcam * n + g] = key;
            idx [(size_t)bcam * n + g] = (unsigned)g;
        }
    }
}

// ---------------------------------------------------------------------------
// Radix sort: histogram -> digit-major scan -> stable scatter (4 passes x B).
// ---------------------------------------------------------------------------
__global__ __launch_bounds__(SORT_TB)
void hist_kernel(const unsigned* __restrict__ keys, unsigned* __restrict__ hist,
                 int n, int numTiles, int shift)
{
    __shared__ unsigned cnt[256];
    int b = blockIdx.y, tile = blockIdx.x;
    for (int i = threadIdx.x; i < 256; i += SORT_TB) cnt[i] = 0;
    __syncthreads();
    const unsigned* k = keys + (size_t)b * n;
    int base = tile * SORT_TILE;
    for (int i = 0; i < SORT_IPT; i++) {
        int p = base + i*SORT_TB + threadIdx.x;
        if (p < n) atomicAdd(&cnt[(k[p] >> shift) & 255u], 1u);
    }
    __syncthreads();
    unsigned* h = hist + (size_t)b * 256 * numTiles;
    for (int d = threadIdx.x; d < 256; d += SORT_TB)
        h[(size_t)d * numTiles + tile] = cnt[d];
}

__global__ __launch_bounds__(SORT_TB)
void scan_kernel(unsigned* __restrict__ hist, int total)  // total = 256*numTiles (multiple of 256)
{
    __shared__ unsigned s[SORT_TB];
    __shared__ unsigned carry;
    unsigned* h = hist + (size_t)blockIdx.x * total;
    if (threadIdx.x == 0) carry = 0;
    __syncthreads();
    for (int base = 0; base < total; base += SORT_TB) {
        unsigned v = h[base + threadIdx.x];
        s[threadIdx.x] = v; __syncthreads();
        for (int off = 1; off < SORT_TB; off <<= 1) {
            unsigned t = (threadIdx.x >= off) ? s[threadIdx.x - off] : 0u;
            __syncthreads();
            s[threadIdx.x] += t;
            __syncthreads();
        }
        unsigned incl = s[threadIdx.x];
        h[base + threadIdx.x] = (incl - v) + carry;
        __syncthreads();
        if (threadIdx.x == SORT_TB - 1) carry += incl;
        __syncthreads();
    }
}

__global__ __launch_bounds__(SORT_TB)
void scatter_kernel(const unsigned* __restrict__ keysIn, const unsigned* __restrict__ idxIn,
                    unsigned* __restrict__ keysOut, unsigned* __restrict__ idxOut,
                    const unsigned* __restrict__ hist, int n, int numTiles, int shift)
{
    __shared__ unsigned waveHist[8][256];
    __shared__ unsigned digitBase[256];
    __shared__ unsigned tileGlobal[256];
    int b = blockIdx.y, tile = blockIdx.x;
    int tid = threadIdx.x, lane = tid & 31, wave = tid >> 5;
    for (int i = tid; i < 256; i += SORT_TB) {
        digitBase[i]  = 0;
        tileGlobal[i] = hist[(size_t)b * 256 * numTiles + (size_t)i * numTiles + tile];
    }
    const unsigned* kin = keysIn + (size_t)b * n;
    const unsigned* iin = idxIn  + (size_t)b * n;
    unsigned* kout = keysOut + (size_t)b * n;
    unsigned* iout = idxOut  + (size_t)b * n;
    int base = tile * SORT_TILE;
    for (int r = 0; r < SORT_IPT; r++) {
        __syncthreads();
        for (int i = tid; i < 8*256; i += SORT_TB) ((unsigned*)waveHist)[i] = 0;
        __syncthreads();
        int pos = base + r*SORT_TB + tid;                 // striped: global order = r*256+tid
        bool valid = (pos < n);
        unsigned key = valid ? kin[pos] : 0xFFFFFFFFu;
        unsigned pay = valid ? iin[pos] : 0u;
        unsigned dig = (key >> shift) & 255u;
        unsigned peers = (unsigned)__ballot(valid);        // wave32 mask
        #pragma unroll
        for (int bit = 0; bit < 8; bit++) {
            unsigned bb = (unsigned)__ballot((dig >> bit) & 1u);
            peers &= ((dig >> bit) & 1u) ? bb : ~bb;
        }
        unsigned lt = (1u << lane) - 1u;
        unsigned waveRank = __popc(peers & lt);
        if (valid && waveRank == 0)                        // one leader per digit per wave
            waveHist[wave][dig] = __popc(peers);
        __syncthreads();
        {   // per-digit sequential scan across the 8 waves (stable order)
            unsigned runn = digitBase[tid];
            #pragma unroll
            for (int w = 0; w < 8; w++) {
                unsigned c = waveHist[w][tid];
                waveHist[w][tid] = runn;
                runn += c;
            }
            digitBase[tid] = runn;
        }
        __syncthreads();
        if (valid) {
            unsigned gpos = tileGlobal[dig] + waveHist[wave][dig] + waveRank;
            kout[gpos] = key;
            iout[gpos] = pay;
        }
    }
}

__global__ __launch_bounds__(256)
void gather_kernel(const float* __restrict__ unsorted, const unsigned* __restrict__ idx,
                   float* __restrict__ out, int n)
{
    int b = blockIdx.y;
    int i = blockIdx.x * blockDim.x + threadIdx.x;
    if (i >= n) return;
    unsigned src = idx[(size_t)b * n + i];
    const float* s = unsorted + ((size_t)b * n + src) * 12;
    float* o = out + ((size_t)b * n + i) * 12;
    #pragma unroll
    for (int j = 0; j < 12; j++) o[j] = s[j];
}

// ---------------------------------------------------------------------------
extern "C" void kernel_launch(void* const* d_in, const int* in_sizes, int n_in,
                              void* d_out, int out_size, void* d_ws, size_t ws_size,
                              hipStream_t stream)
{
    const float* mean  = (const float*)d_in[0];
    const float* vq    = (const float*)d_in[1];
    const float* vs    = (const float*)d_in[2];
    const float* opa   = (const float*)d_in[3];
    const float* color = (const float*)d_in[4];
    const float* P     = (const float*)d_in[5];
    const float* K     = (const float*)d_in[6];
    // d_in[7] = wh (unused by reference math)

    const int N_ = in_sizes[0] / 3;
    const int B_ = in_sizes[5] / 12;
    const int numTiles16 = (N_ + 15) / 16;
    const int numTiles   = (N_ + SORT_TILE - 1) / SORT_TILE;

    auto align256 = [](size_t x) { return (x + 255) & ~(size_t)255; };
    char* ws = (char*)d_ws;
    float*    unsorted = (float*)ws;    ws += align256((size_t)B_ * N_ * 12 * sizeof(float));
    unsigned* keysA    = (unsigned*)ws; ws += align256((size_t)B_ * N_ * sizeof(unsigned));
    unsigned* keysB    = (unsigned*)ws; ws += align256((size_t)B_ * N_ * sizeof(unsigned));
    unsigned* idxA     = (unsigned*)ws; ws += align256((size_t)B_ * N_ * sizeof(unsigned));
    unsigned* idxB     = (unsigned*)ws; ws += align256((size_t)B_ * N_ * sizeof(unsigned));
    unsigned* histBuf  = (unsigned*)ws; ws += align256((size_t)B_ * 256 * numTiles * sizeof(unsigned));

    gs_compute_kernel<<<dim3((numTiles16 + 7) / 8), 256, 0, stream>>>(
        mean, vq, vs, opa, color, P, K, unsorted, keysA, idxA, N_, numTiles16);

    unsigned *kIn = keysA, *iIn = idxA, *kOut = keysB, *iOut = idxB;
    for (int p = 0; p < 4; p++) {
        hist_kernel<<<dim3(numTiles, B_), SORT_TB, 0, stream>>>(kIn, histBuf, N_, numTiles, p * 8);
        scan_kernel<<<dim3(B_), SORT_TB, 0, stream>>>(histBuf, 256 * numTiles);
        scatter_kernel<<<dim3(numTiles, B_), SORT_TB, 0, stream>>>(
            kIn, iIn, kOut, iOut, histBuf, N_, numTiles, p * 8);
        unsigned* t;
        t = kIn; kIn = kOut; kOut = t;
        t = iIn; iIn = iOut; iOut = t;
    }

    gather_kernel<<<dim3((N_ + 255) / 256, B_), 256, 0, stream>>>(
        unsorted, iIn, (float*)d_out, N_);
}